// MultiConditionGNN_67345087201449
// MI455X (gfx1250) — compile-verified
//
#include <hip/hip_runtime.h>

// Problem constants (from reference)
#define B_  4
#define N_  50000
#define D_  128
#define A_  128
#define R_  200
#define E_  600000
#define BN  (B_ * N_)
#define LDW 132   // padded LDS row stride (floats): 132 % 64 banks == 4 -> conflict-free float2 column reads

typedef __attribute__((ext_vector_type(2))) float v2f;
typedef __attribute__((ext_vector_type(8))) float v8f;

__device__ __forceinline__ v8f wmma_f32_4(v2f a, v2f b, v8f c) {
    // D(16x16,f32) += A(16x4,f32) * B(4x16,f32)
    return __builtin_amdgcn_wmma_f32_16x16x4_f32(
        /*neg_a=*/false, a, /*neg_b=*/false, b,
        /*c_mod=*/(short)0, c, /*reuse_a=*/false, /*reuse_b=*/false);
}

// ---------------- scratch init ----------------
__global__ void zero_agg_kernel(float4* agg) {
    size_t i = (size_t)blockIdx.x * blockDim.x + threadIdx.x;   // exactly BN*D/4 threads
    agg[i] = make_float4(0.f, 0.f, 0.f, 0.f);
}

__global__ void zero_recv_kernel(int* recv) {
    int i = blockIdx.x * blockDim.x + threadIdx.x;
    if (i < BN) recv[i] = 0;
}

// ---------------- tiny precomputed projections ----------------
// wr_proj[r][a] = sum_d rela[r][d] * Wr[a][d]
__global__ __launch_bounds__(128) void relproj_kernel(const float* __restrict__ rela,
                                                      const float* __restrict__ Wr,
                                                      float* __restrict__ wr_proj) {
    int r = blockIdx.x, a = threadIdx.x;
    const float* x = rela + r * D_;
    const float* w = Wr + a * D_;
    float s = 0.f;
    #pragma unroll 8
    for (int d = 0; d < D_; ++d) s += x[d] * w[d];
    wr_proj[r * A_ + a] = s;
}

// wq_proj[b][a] = sum_d query[b][d] * Wqr[a][d] + bqr[a]
__global__ __launch_bounds__(128) void qproj_kernel(const float* __restrict__ query,
                                                    const float* __restrict__ Wqr,
                                                    const float* __restrict__ bqr,
                                                    float* __restrict__ wq_proj) {
    int b = blockIdx.x, a = threadIdx.x;
    const float* x = query + b * D_;
    const float* w = Wqr + a * D_;
    float s = bqr[a];
    #pragma unroll 8
    for (int d = 0; d < D_; ++d) s += x[d] * w[d];
    wq_proj[b * A_ + a] = s;
}

// ---------------- edge kernel: attention + weighted scatter ----------------
// 4 waves / block, 16 edges / wave. h_attn = 16x128 @ 128x128 GEMM via WMMA f32 16x16x4.
__global__ __launch_bounds__(128)
void edge_kernel(const int4* __restrict__ edges, const float* __restrict__ hidden,
                 const float* __restrict__ Wsw, const float* __restrict__ wr_proj,
                 const float* __restrict__ wq_proj, const float* __restrict__ wattn,
                 const float* __restrict__ rela, float* __restrict__ agg,
                 int* __restrict__ recv) {
    __shared__ float Ws[128 * LDW];       // Ws_attn_w, row-major, padded
    __shared__ float Xt[4][16 * LDW];     // per-wave mh tile (16 edges x 128)

    const int tid = threadIdx.x, wave = tid >> 5, lane = tid & 31;

    // cooperative load of Ws (once per block)
    for (int i = tid; i < 128 * 32; i += 128) {
        int r = i >> 5, c = (i & 31) * 4;
        *(float4*)&Ws[r * LDW + c] = *(const float4*)(Wsw + r * 128 + c);
    }
    __syncthreads();

    const long tile = (long)blockIdx.x * 4 + wave;
    const long e0   = tile * 16;

    int4 ee = make_int4(0, 0, 0, 0);              // {bat, sub, rel, obj}
    if (lane < 16) ee = edges[e0 + lane];

    // gather mh rows (16 x 128 f32) into LDS, one float4 per lane per row
    for (int m = 0; m < 16; ++m) {
        int row = __shfl(ee.y, m, 32);
        *(float4*)&Xt[wave][m * LDW + lane * 4] =
            *(const float4*)(hidden + (size_t)row * D_ + lane * 4);
    }

    const int lh = lane >> 4, n = lane & 15;

    // C(16 edges x 128 attn) = mh @ Ws^T : 8 N-tiles x 32 K-steps of wmma 16x16x4
    v8f acc[8] = {};
    for (int kk = 0; kk < 32; ++kk) {
        const int k0 = kk * 4 + lh * 2;
        v2f a = *(const v2f*)&Xt[wave][n * LDW + k0];
        #pragma unroll
        for (int nt = 0; nt < 8; ++nt) {
            v2f b = *(const v2f*)&Ws[(nt * 16 + n) * LDW + k0];
            acc[nt] = wmma_f32_4(a, b, acc[nt]);
        }
    }

    float wa[8];
    #pragma unroll
    for (int nt = 0; nt < 8; ++nt) wa[nt] = wattn[nt * 16 + n];

    // per-edge logit: relu(C + wr_proj + wq_proj) . wattn, then sigmoid
    float alpha[8];
    #pragma unroll
    for (int j = 0; j < 8; ++j) {
        const int m = j + 8 * lh;                 // edge index for this lane-half
        const int relm = __shfl(ee.z, m, 32);
        const int batm = __shfl(ee.x, m, 32);
        float sum = 0.f;
        #pragma unroll
        for (int nt = 0; nt < 8; ++nt) {
            const int ai = nt * 16 + n;
            float v = acc[nt][j] + wr_proj[relm * A_ + ai] + wq_proj[batm * A_ + ai];
            v = fmaxf(v, 0.f);
            sum += v * wa[nt];
        }
        sum += __shfl_xor(sum, 1, 32);
        sum += __shfl_xor(sum, 2, 32);
        sum += __shfl_xor(sum, 4, 32);
        sum += __shfl_xor(sum, 8, 32);            // reduce within each 16-lane half
        alpha[j] = 1.f / (1.f + __expf(-sum));
    }

    // message = mh * mr * alpha, scattered with float atomics (L2-resident)
    for (int m = 0; m < 16; ++m) {
        const int obj  = __shfl(ee.w, m, 32);
        const int relm = __shfl(ee.z, m, 32);
        const float al = __shfl(alpha[m & 7], (m >> 3) * 16, 32);
        float4 mh = *(float4*)&Xt[wave][m * LDW + lane * 4];
        float4 mr = *(const float4*)(rela + relm * D_ + lane * 4);
        size_t base = (size_t)obj * D_ + lane * 4;
        atomicAdd(&agg[base + 0], mh.x * mr.x * al);
        atomicAdd(&agg[base + 1], mh.y * mr.y * al);
        atomicAdd(&agg[base + 2], mh.z * mr.z * al);
        atomicAdd(&agg[base + 3], mh.w * mr.w * al);
        if (lane == 0) recv[obj] = 1;             // idempotent flag (deg > 0)
    }
}

// ---------------- node MLP: out = recv ? relu((agg@W1^T+b1)@W2^T+b2) : 0 ----------------
// 2 waves / block, 16 node-rows / wave, both weights LDS-resident.
__global__ __launch_bounds__(64)
void mlp_kernel(const float* __restrict__ agg, const int* __restrict__ recv,
                const float* __restrict__ m1w, const float* __restrict__ m1b,
                const float* __restrict__ m2w, const float* __restrict__ m2b,
                float* __restrict__ out) {
    __shared__ float W1[128 * LDW];
    __shared__ float W2[128 * LDW];
    __shared__ float Xt[2][16 * LDW];

    const int tid = threadIdx.x, wave = tid >> 5, lane = tid & 31;

    for (int i = tid; i < 128 * 32; i += 64) {
        int r = i >> 5, c = (i & 31) * 4;
        *(float4*)&W1[r * LDW + c] = *(const float4*)(m1w + r * 128 + c);
        *(float4*)&W2[r * LDW + c] = *(const float4*)(m2w + r * 128 + c);
    }
    __syncthreads();

    const long row0 = ((long)blockIdx.x * 2 + wave) * 16;

    for (int m = 0; m < 16; ++m)
        *(float4*)&Xt[wave][m * LDW + lane * 4] =
            *(const float4*)(agg + (size_t)(row0 + m) * D_ + lane * 4);

    const int lh = lane >> 4, n = lane & 15;

    v8f acc[8] = {};
    for (int kk = 0; kk < 32; ++kk) {
        const int k0 = kk * 4 + lh * 2;
        v2f a = *(const v2f*)&Xt[wave][n * LDW + k0];
        #pragma unroll
        for (int nt = 0; nt < 8; ++nt) {
            v2f b = *(const v2f*)&W1[(nt * 16 + n) * LDW + k0];
            acc[nt] = wmma_f32_4(a, b, acc[nt]);
        }
    }

    // t1 = acc + b1, relaid out C->A through LDS (per-wave region, no barrier needed)
    #pragma unroll
    for (int nt = 0; nt < 8; ++nt) {
        float bb = m1b[nt * 16 + n];
        #pragma unroll
        for (int j = 0; j < 8; ++j)
            Xt[wave][(j + 8 * lh) * LDW + nt * 16 + n] = acc[nt][j] + bb;
    }

    v8f acc2[8] = {};
    for (int kk = 0; kk < 32; ++kk) {
        const int k0 = kk * 4 + lh * 2;
        v2f a = *(const v2f*)&Xt[wave][n * LDW + k0];
        #pragma unroll
        for (int nt = 0; nt < 8; ++nt) {
            v2f b = *(const v2f*)&W2[(nt * 16 + n) * LDW + k0];
            acc2[nt] = wmma_f32_4(a, b, acc2[nt]);
        }
    }

    #pragma unroll
    for (int j = 0; j < 8; ++j) {
        const long r = row0 + j + 8 * lh;
        const int rv = recv[r];
        #pragma unroll
        for (int nt = 0; nt < 8; ++nt) {
            float v = fmaxf(acc2[nt][j] + m2b[nt * 16 + n], 0.f);
            out[(size_t)r * D_ + nt * 16 + n] = rv ? v : 0.f;
        }
    }
}

// ---------------- host launcher ----------------
extern "C" void kernel_launch(void* const* d_in, const int* in_sizes, int n_in,
                              void* d_out, int out_size, void* d_ws, size_t ws_size,
                              hipStream_t stream) {
    const float* query = (const float*)d_in[0];
    // d_in[1] q_sub, d_in[2] q_rel: unused by reference
    const float* hidden = (const float*)d_in[3];
    const int4*  edges  = (const int4*)d_in[4];
    // d_in[5] nodes: unused
    const float* Wsw   = (const float*)d_in[6];
    const float* Wrw   = (const float*)d_in[7];
    const float* Wqrw  = (const float*)d_in[8];
    const float* Wqrb  = (const float*)d_in[9];
    const float* Wattn = (const float*)d_in[10];
    const float* rela  = (const float*)d_in[11];
    const float* m1w   = (const float*)d_in[12];
    const float* m1b   = (const float*)d_in[13];
    const float* m2w   = (const float*)d_in[14];
    const float* m2b   = (const float*)d_in[15];
    float* out = (float*)d_out;

    // workspace carve-up
    char*  ws      = (char*)d_ws;
    float* agg     = (float*)ws;                                   // BN*D f32   (102.4 MB)
    int*   recv    = (int*)(ws + (size_t)BN * D_ * 4);             // BN i32
    float* wr_proj = (float*)(ws + (size_t)BN * D_ * 4 + (size_t)BN * 4);  // R*A f32
    float* wq_proj = wr_proj + R_ * A_;                            // B*A f32

    zero_agg_kernel<<<(BN * (D_ / 4)) / 256, 256, 0, stream>>>((float4*)agg);
    zero_recv_kernel<<<(BN + 255) / 256, 256, 0, stream>>>(recv);
    relproj_kernel<<<R_, 128, 0, stream>>>(rela, Wrw, wr_proj);
    qproj_kernel<<<B_, 128, 0, stream>>>(query, Wqrw, Wqrb, wq_proj);
    edge_kernel<<<E_ / 64, 128, 0, stream>>>(edges, hidden, Wsw, wr_proj, wq_proj,
                                             Wattn, rela, agg, recv);
    mlp_kernel<<<BN / 32, 64, 0, stream>>>(agg, recv, m1w, m1b, m2w, m2b, out);
}